// MAE_bbox_weighted_66967130079244
// MI455X (gfx1250) — compile-verified
//
#include <hip/hip_runtime.h>

#define ALPHA   100.0f
#define MARG    5
#define IMG_H   512
#define IMG_W   512
#define CB      96                       // B*C = 32*3
#define VEC_PER_IMG 65536                // H*W/4
#define GRIDX   32
#define CHUNK   (VEC_PER_IMG / GRIDX)    // 2048 float4 per block
#define ITERS   (CHUNK / 256)            // 8 iterations per thread
#define NBLOCKS (CB * GRIDX)             // 3072 partials
#define INV_N   (1.0f / 25165824.0f)     // 1 / (32*3*512*512)

typedef float v2f __attribute__((ext_vector_type(2)));
typedef float v8f __attribute__((ext_vector_type(8)));

// Reduce 32 lane partials to a single wave total using V_WMMA_F32_16X16X4_F32.
// A-layout (16x4 f32): lane L holds A[L%16, K] with K={0,1} for lanes 0-15 and
// K={2,3} for lanes 16-31 in VGPRs {0,1}. We put the partial in VGPR0, 0 in
// VGPR1, and B = all-ones, so D[m,n] = p[m] + p[m+16] for every n.
// D-layout: VGPR r holds D[r, n] in lanes 0-15 and D[r+8, n] in lanes 16-31.
// Summing the 8 D registers gives sum(p[0..7]+p[16..23]) in lanes 0-15 and
// sum(p[8..15]+p[24..31]) in lanes 16-31; one xor-16 shuffle completes it.
__device__ __forceinline__ float wave_sum32(float p) {
  v2f a; a[0] = p;    a[1] = 0.0f;
  v2f b; b[0] = 1.0f; b[1] = 1.0f;
  v8f c = {};
  v8f d = __builtin_amdgcn_wmma_f32_16x16x4_f32(
      /*neg_a=*/false, a, /*neg_b=*/false, b,
      /*c_mod=*/(short)0, c, /*reuse_a=*/false, /*reuse_b=*/false);
  float t = ((d[0] + d[1]) + (d[2] + d[3])) + ((d[4] + d[5]) + (d[6] + d[7]));
  t += __shfl_xor(t, 16, 32);
  return t;
}

// Ramp factor matching the reference's jnp.where chain (evaluated in order).
__device__ __forceinline__ int ramp(int p, int lo, int hi) {
  if (p >= lo && p < hi)               return MARG;               // inside span
  if (p >= lo - MARG && p < lo)        return p - lo + MARG + 1;  // leading strip 1..m
  if (p >= hi && p < hi + MARG)        return MARG - (p - hi);    // trailing strip m..1
  return 0;
}

__device__ __forceinline__ float pix_weight(int fx, int fy, bool in_x, bool in_y,
                                            bool cond) {
  bool inside = in_x && in_y;
  float w = inside ? ALPHA : 1.0f;
  bool band = (fy > 0) && (fx > 0) && !inside;
  if (cond && band) {
    int f = fx < fy ? fx : fy;
    w = (float)f * (ALPHA / (float)MARG);
  }
  return w;
}

__global__ void __launch_bounds__(256)
mae_bbox_partial(const float* __restrict__ x, const float* __restrict__ y,
                 const int* __restrict__ bbox, float* __restrict__ partial) {
  const int bc = blockIdx.y;        // b*3 + c
  const int b  = bc / 3;

  // bbox[b] = 4 points of (x, y); uniform per block -> scalar loads.
  const int* bb = bbox + b * 8;
  const int px0 = bb[0], py0 = bb[1], px1 = bb[2], py1 = bb[3];
  const int px2 = bb[4], py2 = bb[5], px3 = bb[6], py3 = bb[7];
  const int xmin = min(min(px0, px1), min(px2, px3));
  const int xmax = max(max(px0, px1), max(px2, px3));
  const int ymin = min(min(py0, py1), min(py2, py3));
  const int ymax = max(max(py0, py1), max(py2, py3));
  const bool cond = (xmin - MARG > 0) && (xmax + MARG < IMG_W) &&
                    (ymin - MARG > 0) && (ymax + MARG < IMG_H);

  const float4* __restrict__ xp = (const float4*)x + (size_t)bc * VEC_PER_IMG;
  const float4* __restrict__ yp = (const float4*)y + (size_t)bc * VEC_PER_IMG;

  float acc = 0.0f;
#pragma unroll
  for (int it = 0; it < ITERS; ++it) {
    const int v = blockIdx.x * CHUNK + it * 256 + threadIdx.x;
    const float4 xv = xp[v];   // global_load_b128
    const float4 yv = yp[v];
    const int row = v >> 7;            // W/4 = 128 float4 per row
    const int col = (v & 127) << 2;    // leading x of this float4

    const bool in_y = (row >= ymin) && (row < ymax);
    const int  fy   = ramp(row, ymin, ymax);

    const float dx[4] = {xv.x - yv.x, xv.y - yv.y, xv.z - yv.z, xv.w - yv.w};
#pragma unroll
    for (int k = 0; k < 4; ++k) {
      const int  xs   = col + k;
      const bool in_x = (xs >= xmin) && (xs < xmax);
      const int  fx   = ramp(xs, xmin, xmax);
      acc += fabsf(dx[k]) * pix_weight(fx, fy, in_x, in_y, cond);
    }
  }

  // Wave reduction via WMMA (EXEC all-1s here: no divergence yet).
  const float t = wave_sum32(acc);

  __shared__ float ws[8];
  const int lane = threadIdx.x & 31;
  const int wv   = threadIdx.x >> 5;
  if (lane == 0) ws[wv] = t;
  __syncthreads();
  if (threadIdx.x == 0) {
    float s = 0.0f;
#pragma unroll
    for (int i = 0; i < 8; ++i) s += ws[i];
    partial[blockIdx.y * GRIDX + blockIdx.x] = s;
  }
}

__global__ void __launch_bounds__(256)
mae_bbox_final(const float* __restrict__ partial, float* __restrict__ out) {
  float acc = 0.0f;
#pragma unroll
  for (int i = 0; i < NBLOCKS / 256; ++i)   // 12 exact iterations, EXEC stays full
    acc += partial[i * 256 + threadIdx.x];

  const float t = wave_sum32(acc);

  __shared__ float ws[8];
  const int lane = threadIdx.x & 31;
  const int wv   = threadIdx.x >> 5;
  if (lane == 0) ws[wv] = t;
  __syncthreads();
  if (threadIdx.x == 0) {
    float s = 0.0f;
#pragma unroll
    for (int i = 0; i < 8; ++i) s += ws[i];
    out[0] = s * INV_N;
  }
}

extern "C" void kernel_launch(void* const* d_in, const int* in_sizes, int n_in,
                              void* d_out, int out_size, void* d_ws, size_t ws_size,
                              hipStream_t stream) {
  const float* x    = (const float*)d_in[0];
  const float* y    = (const float*)d_in[1];
  const int*   bbox = (const int*)d_in[2];
  float* partial = (float*)d_ws;          // 3072 floats = 12 KB scratch

  dim3 grid(GRIDX, CB);
  mae_bbox_partial<<<grid, 256, 0, stream>>>(x, y, bbox, partial);
  mae_bbox_final<<<1, 256, 0, stream>>>(partial, (float*)d_out);
}